// NN_split_14516989460965
// MI455X (gfx1250) — compile-verified
//
#include <hip/hip_runtime.h>
#include <math.h>

// Problem constants (match reference)
#define BATCH   1024
#define NGENES  20000
#define HID     2048
#define GRP     64
#define HCHUNK  4        // h's per block in hidden_kernel (each thread owns 4 b's)
#define OUTWAVES 4       // waves per block in out_kernel (K-split)

typedef __attribute__((ext_vector_type(2))) float v2f;
typedef __attribute__((ext_vector_type(8))) float v8f;

// ---------------------------------------------------------------------------
// Kernel 1: transpose x (BATCH x NGENES) -> xT (NGENES x BATCH)
// so that gathers at a fixed gene index are contiguous over b. x (80 MB) and
// xT both fit in the 192 MB L2, so all later gather traffic is L2 hits.
// ---------------------------------------------------------------------------
__global__ void transpose_kernel(const float* __restrict__ x,
                                 float* __restrict__ xT) {
    __shared__ float tile[32][33];
    const int jBase = blockIdx.x * 32;   // gene tile
    const int bBase = blockIdx.y * 32;   // batch tile
    const int tx = threadIdx.x;          // 0..31
    const int ty = threadIdx.y;          // 0..7
#pragma unroll
    for (int i = 0; i < 32; i += 8) {
        tile[ty + i][tx] = x[(size_t)(bBase + ty + i) * NGENES + (jBase + tx)];
    }
    __syncthreads();
#pragma unroll
    for (int i = 0; i < 32; i += 8) {
        xT[(size_t)(jBase + ty + i) * BATCH + (bBase + tx)] = tile[tx][ty + i];
    }
}

// ---------------------------------------------------------------------------
// Kernel 2: hidden[h][b] = relu( sum_g xT[idx[h,g]][b] * w[h,g] + gb[h] )
// Each thread owns 4 consecutive b's -> one global_load_b128 per gathered
// element group (4x fewer load issues than b32, still fully coalesced:
// 32 lanes x 16B = 4 cachelines per wave, served from L2-resident xT).
// Indices/weights for the block's HCHUNK h's staged in LDS (lane-uniform).
// hidden stored [H][B] so b128 stores are coalesced.
// ---------------------------------------------------------------------------
__global__ void hidden_kernel(const float* __restrict__ xT,
                              const int*   __restrict__ gidx,
                              const float* __restrict__ gw,
                              const float* __restrict__ gb,
                              float*       __restrict__ hidden) {
    __shared__ int   s_idx[HCHUNK * GRP];
    __shared__ float s_w[HCHUNK * GRP];

    const int h0  = blockIdx.x * HCHUNK;
    const int tid = threadIdx.x;                 // 0..255, owns b = 4*tid .. 4*tid+3

    for (int t = tid; t < HCHUNK * GRP; t += blockDim.x) {
        s_idx[t] = gidx[(size_t)h0 * GRP + t];
        s_w[t]   = gw[(size_t)h0 * GRP + t];
    }
    __syncthreads();

    const float4* xT4 = (const float4*)xT;       // row stride BATCH/4 = 256

    for (int hh = 0; hh < HCHUNK; ++hh) {
        float4 acc = make_float4(0.f, 0.f, 0.f, 0.f);
#pragma unroll 8
        for (int g = 0; g < GRP; ++g) {
            const int   j = s_idx[hh * GRP + g];
            const float w = s_w[hh * GRP + g];
            const float4 v = xT4[(size_t)j * (BATCH / 4) + tid];
            acc.x = fmaf(v.x, w, acc.x);
            acc.y = fmaf(v.y, w, acc.y);
            acc.z = fmaf(v.z, w, acc.z);
            acc.w = fmaf(v.w, w, acc.w);
        }
        const float bias = gb[h0 + hh];
        float4 hv;
        hv.x = fmaxf(acc.x + bias, 0.f);
        hv.y = fmaxf(acc.y + bias, 0.f);
        hv.z = fmaxf(acc.z + bias, 0.f);
        hv.w = fmaxf(acc.w + bias, 0.f);
        ((float4*)hidden)[(size_t)(h0 + hh) * (BATCH / 4) + tid] = hv;
    }
}

// ---------------------------------------------------------------------------
// Kernel 3: out[b] = sigmoid( sum_h hidden[h][b] * w2[h] + b2 )
// GEMV via V_WMMA_F32_16X16X4_F32 (pure f32 -> reference accuracy).
// 4 waves per block, K=2048 split into 4 segments of 512 (128 WMMA steps
// each) for latency hiding; deterministic LDS reduction of D column 0.
// Per K-step of 4 h's:
//   A (16x4): lane L<16 holds (hidden[k0+0][b0+L], hidden[k0+1][b0+L]),
//             lane L>=16 holds rows k0+2 / k0+3 (K-split VGPR layout).
//   B (4x16): only column 0 nonzero -> lane 0 = (w2[k0], w2[k0+1]),
//             lane 16 = (w2[k0+2], w2[k0+3]), all other lanes 0.
// Column 0 of D lives in lanes 0 (M=0..7) and 16 (M=8..15) across 8 VGPRs.
// ---------------------------------------------------------------------------
__global__ void out_kernel(const float* __restrict__ hidden,
                           const float* __restrict__ w2,
                           const float* __restrict__ b2,
                           float*       __restrict__ out) {
    __shared__ float red[OUTWAVES][16];

    const int wave = threadIdx.x >> 5;       // 0..3
    const int lane = threadIdx.x & 31;
    const int b0   = blockIdx.x * 16;        // 16-batch tile
    const int m    = lane & 15;
    const int koff = (lane < 16) ? 0 : 2;
    const bool sel = (m == 0);               // lanes 0 and 16 carry B data

    const int kBeg = wave * (HID / OUTWAVES);
    const int kEnd = kBeg + (HID / OUTWAVES);

    v8f c = {};
#pragma unroll 4
    for (int k0 = kBeg; k0 < kEnd; k0 += 4) {
        const float* hrow = hidden + (size_t)(k0 + koff) * BATCH + b0 + m;
        v2f a;
        a.x = hrow[0];
        a.y = hrow[BATCH];

        const float w0 = w2[k0 + koff];
        const float w1 = w2[k0 + koff + 1];
        v2f bf;
        bf.x = sel ? w0 : 0.0f;
        bf.y = sel ? w1 : 0.0f;

        // 8 args: (neg_a, A, neg_b, B, c_mod, C, reuse_a, reuse_b)
        c = __builtin_amdgcn_wmma_f32_16x16x4_f32(
                false, a, false, bf, (short)0, c, false, false);
    }

    // Extract column 0 of this wave's partial D into LDS.
    if (lane == 0) {
#pragma unroll
        for (int r = 0; r < 8; ++r) red[wave][r] = c[r];
    }
    if (lane == 16) {
#pragma unroll
        for (int r = 0; r < 8; ++r) red[wave][8 + r] = c[r];
    }
    __syncthreads();

    // Deterministic cross-wave reduction + bias + sigmoid.
    if (threadIdx.x < 16) {
        float z = b2[0];
#pragma unroll
        for (int wv = 0; wv < OUTWAVES; ++wv) z += red[wv][threadIdx.x];
        out[b0 + threadIdx.x] = 1.0f / (1.0f + expf(-z));
    }
}

// ---------------------------------------------------------------------------
// Host launcher
// ---------------------------------------------------------------------------
extern "C" void kernel_launch(void* const* d_in, const int* in_sizes, int n_in,
                              void* d_out, int out_size, void* d_ws, size_t ws_size,
                              hipStream_t stream) {
    const float* x    = (const float*)d_in[0];   // (B, NGENES) f32
    const int*   gidx = (const int*)  d_in[1];   // (H, G) int
    const float* gw   = (const float*)d_in[2];   // (H, G) f32
    const float* gb   = (const float*)d_in[3];   // (H,)   f32
    const float* w2   = (const float*)d_in[4];   // (1, H) f32
    const float* b2   = (const float*)d_in[5];   // (1,)   f32
    float* out = (float*)d_out;                  // (B, 1) f32

    char*  ws     = (char*)d_ws;
    float* xT     = (float*)ws;                                            // NGENES*BATCH f32
    float* hidden = (float*)(ws + (size_t)NGENES * BATCH * sizeof(float)); // HID*BATCH f32

    // 1) transpose x -> xT   (grid: 625 x 32 tiles of 32x32)
    dim3 tBlk(32, 8);
    dim3 tGrd(NGENES / 32, BATCH / 32);
    transpose_kernel<<<tGrd, tBlk, 0, stream>>>(x, xT);

    // 2) hidden  (512 blocks of 256 threads; each thread owns 4 b's)
    hidden_kernel<<<HID / HCHUNK, 256, 0, stream>>>(xT, gidx, gw, gb, hidden);

    // 3) WMMA GEMV + sigmoid (64 blocks x 4 waves, K split across waves)
    out_kernel<<<BATCH / 16, 32 * OUTWAVES, 0, stream>>>(hidden, w2, b2, out);
}